// DEHNN_layer_15590731285076
// MI455X (gfx1250) — compile-verified
//
#include <hip/hip_runtime.h>

// ---------------------------------------------------------------------------
// DEHNN layer on gfx1250 (MI455X): fp32 end-to-end.
//   - GEMMs via V_WMMA_F32_16X16X4_F32 (fp32 matrix pipe, wave32)
//   - A tiles staged in LDS via global_load_async_to_lds_b128 (ASYNCcnt)
//   - segment ops via wave-per-edge float4 gathers + global_atomic_add_f32
// Memory-bound workload (~0.8 GB edge traffic, all tensors fit 192 MB L2).
// ---------------------------------------------------------------------------

#define D   128
#define DV4 32   // D / 4 floats per float4 row chunk

#define ROW_TILES      4                       // 16-row WMMA tiles per block
#define ROWS_PER_BLOCK (ROW_TILES * 16)        // 64
#define LDS_STRIDE     132                     // 128 + 4 pad: 16B-aligned rows, no bank conflicts

typedef __attribute__((ext_vector_type(2))) float v2f;
typedef __attribute__((ext_vector_type(8))) float v8f;

// ---------------- utility: zero a float buffer (grid-stride) ----------------
__global__ void zero_f32(float* __restrict__ p, int n) {
  int i = blockIdx.x * blockDim.x + threadIdx.x;
  int stride = gridDim.x * blockDim.x;
  for (; i < n; i += stride) p[i] = 0.0f;
}

// ------------- segment sum + count: one wave per edge, float4 lanes ---------
__global__ void scatter_sum_cnt(const float* __restrict__ feat,
                                const int* __restrict__ src,
                                const int* __restrict__ dst,
                                float* __restrict__ sum,
                                float* __restrict__ cnt, int E) {
  int t = blockIdx.x * blockDim.x + threadIdx.x;
  int e = t >> 5;
  if (e >= E) return;
  int lane = t & 31;
  int s = src[e], d = dst[e];
  float4 v = ((const float4*)feat)[(size_t)s * DV4 + lane];
  float* o = sum + (size_t)d * D + lane * 4;
  atomicAdd(o + 0, v.x);
  atomicAdd(o + 1, v.y);
  atomicAdd(o + 2, v.z);
  atomicAdd(o + 3, v.w);
  if (lane == 0) atomicAdd(cnt + d, 1.0f);
}

// ---------------- GCN degree count (thread per edge) ------------------------
__global__ void deg_count(const int* __restrict__ dst, float* __restrict__ deg, int E) {
  int e = blockIdx.x * blockDim.x + threadIdx.x;
  if (e < E) atomicAdd(deg + dst[e], 1.0f);
}

__global__ void rsqrt_deg(const float* __restrict__ deg, float* __restrict__ dinv, int n) {
  int i = blockIdx.x * blockDim.x + threadIdx.x;
  if (i < n) dinv[i] = rsqrtf(deg[i] + 1.0f);  // deg includes +1 self-loop
}

// ------------- GCN normalized scatter: x2[d] += dinv[s]*dinv[d]*h[s] --------
__global__ void gcn_scatter(const float* __restrict__ h,
                            const int* __restrict__ src,
                            const int* __restrict__ dst,
                            const float* __restrict__ dinv,
                            float* __restrict__ out, int E) {
  int t = blockIdx.x * blockDim.x + threadIdx.x;
  int e = t >> 5;
  if (e >= E) return;
  int lane = t & 31;
  int s = src[e], d = dst[e];
  float norm = dinv[s] * dinv[d];
  float4 v = ((const float4*)h)[(size_t)s * DV4 + lane];
  float* o = out + (size_t)d * D + lane * 4;
  atomicAdd(o + 0, norm * v.x);
  atomicAdd(o + 1, norm * v.y);
  atomicAdd(o + 2, norm * v.z);
  atomicAdd(o + 3, norm * v.w);
}

// ------------- self-loop term: x2 += h * dinv^2 + b_gcn ---------------------
__global__ void gcn_selfloop(float* __restrict__ x2, const float* __restrict__ h,
                             const float* __restrict__ dinv,
                             const float* __restrict__ bias, int n) {
  int i = blockIdx.x * blockDim.x + threadIdx.x;
  int stride = gridDim.x * blockDim.x;
  for (; i < n; i += stride) {
    int row = i >> 7;      // / 128
    int c   = i & 127;     // % 128
    float di = dinv[row];
    x2[i] += h[i] * di * di + bias[c];
  }
}

// ---------------------------------------------------------------------------
// Fused GEMM: out[r,c] = act( resid[r,c] + (A[r,:]*inv(cnt[r])) @ W[:,c] + b[c] )
// Block = 256 threads = 8 waves, covering 64 rows x 128 cols.
//   1) async-copy the 64x128 fp32 A tile into LDS (global_load_async_to_lds_b128,
//      drained with s_wait_asynccnt, then workgroup barrier)
//   2) wave w owns columns [16w,16w+16); per K-step it loads one B fragment
//      from global (L2-resident W) and reuses it across 4 row-tile WMMAs,
//      with A fragments read from LDS (ds_load_b64, conflict-free via pad).
// Fragment layouts per CDNA5 ISA 7.12.2 (fp32, wave32):
//   A 16x4:  lane<16 -> (M=lane, K=k0+{0,1}); lane>=16 -> (M=lane-16, K=k0+{2,3})
//   B 4x16:  VGPR0 lanes0-15 K=k0 N=lane, lanes16-31 K=k0+2; VGPR1 -> K=k0+1/k0+3
//   C 16x16: VGPR i: lanes0-15 (M=i, N=lane); lanes16-31 (M=i+8, N=lane-16)
// Tail block: load rows clamped, stores predicated; WMMAs stay uniform (EXEC all-1s).
// ---------------------------------------------------------------------------
__global__ __launch_bounds__(256) void gemm128_wmma(
    const float* __restrict__ A, const float* __restrict__ cnt,
    const float* __restrict__ W, const float* __restrict__ bias,
    const float* __restrict__ resid, float* __restrict__ out,
    int nrows, int relu) {
  __shared__ float tileA[ROWS_PER_BLOCK * LDS_STRIDE];

  const int tid = threadIdx.x;
  const int rowbase = blockIdx.x * ROWS_PER_BLOCK;

  // ---- stage A tile (64 rows x 128 f32 = 32 KB) into LDS: 8 b128 chunks/thread
  {
    const unsigned lds_base = (unsigned)(unsigned long long)tileA;
#pragma unroll
    for (int i = 0; i < 8; i++) {
      int c   = tid + i * 256;         // chunk id 0..2047 (16 B each)
      int row = c >> 5;                // 0..63
      int cc  = c & 31;                // 0..31 (x4 floats)
      int grow = rowbase + row;
      if (grow >= nrows) grow = nrows - 1;          // clamp for tail block
      const float* g = A + (size_t)grow * D + cc * 4;
      unsigned l = lds_base + (unsigned)(row * LDS_STRIDE + cc * 4) * 4u;
      asm volatile("global_load_async_to_lds_b128 %0, %1, off"
                   :: "v"(l), "v"(g) : "memory");
    }
#if __has_builtin(__builtin_amdgcn_s_wait_asynccnt)
    __builtin_amdgcn_s_wait_asynccnt(0);
#else
    asm volatile("s_wait_asynccnt 0" ::: "memory");
#endif
  }
  __syncthreads();

  const int lane = tid & 31;
  const int wave = tid >> 5;
  const int n0   = wave << 4;            // column stripe base
  const int r15  = lane & 15;
  const int half = lane >> 4;            // 0 or 1
  const int col  = n0 + r15;

  // per-row-tile inverse counts (segment mean folded into A fragment loads)
  float inv[ROW_TILES];
#pragma unroll
  for (int t = 0; t < ROW_TILES; t++) {
    inv[t] = 1.0f;
    if (cnt) {
      int r = rowbase + t * 16 + r15;
      if (r >= nrows) r = nrows - 1;
      inv[t] = 1.0f / fmaxf(cnt[r], 1.0f);
    }
  }

  v8f acc[ROW_TILES];
#pragma unroll
  for (int t = 0; t < ROW_TILES; t++)
    acc[t] = (v8f){0.f, 0.f, 0.f, 0.f, 0.f, 0.f, 0.f, 0.f};

#pragma unroll
  for (int k0 = 0; k0 < D; k0 += 4) {
    const int ka = k0 + half * 2;
    v2f b;
    b.x = W[(size_t)ka * D + col];
    b.y = W[(size_t)(ka + 1) * D + col];
#pragma unroll
    for (int t = 0; t < ROW_TILES; t++) {
      const float2 av = *(const float2*)&tileA[(t * 16 + r15) * LDS_STRIDE + ka];
      v2f a;
      a.x = av.x * inv[t];
      a.y = av.y * inv[t];
      acc[t] = __builtin_amdgcn_wmma_f32_16x16x4_f32(
          /*neg_a=*/false, a, /*neg_b=*/false, b,
          /*c_mod=*/(short)0, acc[t], /*reuse_a=*/false, /*reuse_b=*/false);
    }
  }

  const float bb = bias ? bias[col] : 0.0f;
#pragma unroll
  for (int t = 0; t < ROW_TILES; t++) {
#pragma unroll
    for (int i = 0; i < 8; i++) {
      const int r = rowbase + t * 16 + i + half * 8;
      if (r < nrows) {
        float v = acc[t][i] + bb;
        if (resid) v += resid[(size_t)r * D + col];
        if (relu)  v = fmaxf(v, 0.0f);
        out[(size_t)r * D + col] = v;
      }
    }
  }
}

// ---------------------------------------------------------------------------
extern "C" void kernel_launch(void* const* d_in, const int* in_sizes, int n_in,
                              void* d_out, int out_size, void* d_ws, size_t ws_size,
                              hipStream_t stream) {
  const float* emb = (const float*)d_in[0];
  const float* Wd  = (const float*)d_in[1];
  const float* bd  = (const float*)d_in[2];
  const float* Wg  = (const float*)d_in[3];
  const float* bg  = (const float*)d_in[4];
  const float* Wu  = (const float*)d_in[5];
  const float* bu  = (const float*)d_in[6];
  const int* d2u = (const int*)d_in[7];
  const int* sl  = (const int*)d_in[8];
  const int* u2d = (const int*)d_in[9];

  const int N  = in_sizes[0] / D;       // 100000
  const int E1 = in_sizes[7] / 2;
  const int E2 = in_sizes[8] / 2;
  const int E3 = in_sizes[9] / 2;
  const size_t nd = (size_t)N * D;

  // Workspace layout (floats): 3 node-feature buffers + 3 per-node scalars.
  float* ws   = (float*)d_ws;
  float* sumA = ws;             // [N,D] segment sums (reused for stage 3)
  float* bufX = sumA + nd;      // [N,D] x, then reused as x2 accumulator
  float* bufH = bufX + nd;      // [N,D] h = x @ W_gcn
  float* cnt  = bufH + nd;      // [N]
  float* deg  = cnt + N;        // [N]
  float* dinv = deg + N;        // [N]

  auto cdiv = [](long a, long b) { return (int)((a + b - 1) / b); };
  const int ZB = 8192;          // grid-stride zero blocks
  const int GB = cdiv(N, ROWS_PER_BLOCK);  // GEMM blocks (64 rows each)

  // ---- Stage 1: down2up segment mean + W_down + residual + relu ----
  zero_f32<<<ZB, 256, 0, stream>>>(sumA, (int)nd);
  zero_f32<<<cdiv(N, 256), 256, 0, stream>>>(cnt, N);
  scatter_sum_cnt<<<cdiv((long)E1 * 32, 256), 256, 0, stream>>>(
      emb, d2u, d2u + E1, sumA, cnt, E1);
  gemm128_wmma<<<GB, 256, 0, stream>>>(sumA, cnt, Wd, bd, emb, bufX, N, 1);

  // ---- Stage 2: GCN ----
  gemm128_wmma<<<GB, 256, 0, stream>>>(bufX, nullptr, Wg, nullptr, nullptr, bufH, N, 0);
  zero_f32<<<cdiv(N, 256), 256, 0, stream>>>(deg, N);
  deg_count<<<cdiv(E2, 256), 256, 0, stream>>>(sl + E2, deg, E2);
  rsqrt_deg<<<cdiv(N, 256), 256, 0, stream>>>(deg, dinv, N);
  zero_f32<<<ZB, 256, 0, stream>>>(bufX, (int)nd);   // bufX becomes x2
  gcn_scatter<<<cdiv((long)E2 * 32, 256), 256, 0, stream>>>(
      bufH, sl, sl + E2, dinv, bufX, E2);
  gcn_selfloop<<<ZB, 256, 0, stream>>>(bufX, bufH, dinv, bg, (int)nd);

  // ---- Stage 3: up2down segment mean + W_up + residual + relu -> d_out ----
  zero_f32<<<ZB, 256, 0, stream>>>(sumA, (int)nd);
  zero_f32<<<cdiv(N, 256), 256, 0, stream>>>(cnt, N);
  scatter_sum_cnt<<<cdiv((long)E3 * 32, 256), 256, 0, stream>>>(
      bufX, u2d, u2d + E3, sumA, cnt, E3);
  gemm128_wmma<<<GB, 256, 0, stream>>>(sumA, cnt, Wu, bu, bufX, (float*)d_out, N, 1);
}